// DGLGCN__37709812859404
// MI455X (gfx1250) — compile-verified
//
#include <hip/hip_runtime.h>
#include <math.h>

// ---------------- types for WMMA fragments ----------------
typedef float v2f __attribute__((ext_vector_type(2)));
typedef float v8f __attribute__((ext_vector_type(8)));

// ---------------- async-to-LDS pointer types (from hipcc diagnostic) -------
typedef int v4i __attribute__((__vector_size__(16)));
typedef __attribute__((address_space(1))) v4i gv4i;   // global b128
typedef __attribute__((address_space(3))) v4i lv4i;   // LDS b128

#define D_HID 128   // D_IN == D_H == 128 in the reference
#define NT    4     // 16-col output tiles per wave (64 cols per block)

#if __has_builtin(__builtin_amdgcn_global_load_async_to_lds_b128) && \
    __has_builtin(__builtin_amdgcn_s_wait_asynccnt)
#define USE_ASYNC_LDS 1
#else
#define USE_ASYNC_LDS 0
#endif

// ---------------------------------------------------------------------------
// 1) CSR row pointer from sorted dst: row_ptr[n] = lower_bound(dst, n)
// ---------------------------------------------------------------------------
__global__ void rowptr_kernel(const int* __restrict__ dst, int E,
                              int* __restrict__ rp, int N) {
    int n = blockIdx.x * blockDim.x + threadIdx.x;
    if (n > N) return;
    int lo = 0, hi = E;
    while (lo < hi) {
        int mid = (lo + hi) >> 1;
        if (dst[mid] < n) lo = mid + 1; else hi = mid;
    }
    rp[n] = lo;
}

// ---------------------------------------------------------------------------
// 2) SpMM: agg[n,:] = sum_{e in row n} edge_w[e] * h[src[e],:]
//    One wave per node; lane owns 4 contiguous floats (float4, coalesced).
//    Sequential edge loop per node -> deterministic.
// ---------------------------------------------------------------------------
__global__ void spmm_kernel(const float4* __restrict__ h,   // [N,32] float4
                            const float*  __restrict__ ew,  // [E]
                            const int*    __restrict__ src, // [E]
                            const int*    __restrict__ rp,  // [N+1]
                            float4* __restrict__ agg,       // [N,32] float4
                            int N) {
    int wave = threadIdx.x >> 5;
    int lane = threadIdx.x & 31;
    int node = blockIdx.x * (blockDim.x >> 5) + wave;
    if (node >= N) return;
    int e0 = rp[node], e1 = rp[node + 1];
    float ax = 0.f, ay = 0.f, az = 0.f, aw = 0.f;
    for (int e = e0; e < e1; ++e) {
        int   s = src[e];
        float w = ew[e];
        float4 v = h[(size_t)s * 32 + lane];
        ax = fmaf(w, v.x, ax);
        ay = fmaf(w, v.y, ay);
        az = fmaf(w, v.z, az);
        aw = fmaf(w, v.w, aw);
    }
    float4 r; r.x = ax; r.y = ay; r.z = az; r.w = aw;
    agg[(size_t)node * 32 + lane] = r;
}

// ---------------------------------------------------------------------------
// 3) GEMM via V_WMMA_F32_16X16X4_F32: out = act(A[N,128] @ W[128,Dout] + b)
//    Block = 256 threads = 8 waves. Block stages W[:, n0:n0+64] in LDS
//    (async-to-LDS b128 on gfx1250 when tile is full width). Each wave
//    computes a 16x64 output strip: one A-frag feeds NT=4 WMMAs -> 4x
//    A-operand reuse, 128 wmma ops per wave.
//    Fragment layouts (ISA 7.12.2, f32):
//      A 16x4 : lane l -> row l%16, a[v] = A[row][k + 2*(l/16) + v]
//      B 4x16 : b[v]  = W[k + 2*(l/16) + v][col]
//      C 16x16: c[r] -> (M = 8*(l/16) + r, col)
// ---------------------------------------------------------------------------
__global__ __launch_bounds__(256)
void gemm_wmma_kernel(const float* __restrict__ A,     // [N,128]
                      const float* __restrict__ W,     // [128,Dout] row-major
                      const float* __restrict__ bias,  // [Dout]
                      float* __restrict__ out,         // [N,Dout]
                      int N, int Dout, int doRelu) {
    __shared__ float Wl[D_HID * 64];  // 32 KB: W[:, n0:n0+64], stride 64
    __shared__ float bl[64];

    const int tid = threadIdx.x;
    const int n0  = blockIdx.y * 64;

#if USE_ASYNC_LDS
    if (((Dout & 63) == 0)) {
        // gfx1250 async copy global->LDS, tracked with ASYNCcnt.
        // 128 rows x 16 float4 per row = 2048 b128 transfers, 8 per lane.
        for (int idx = tid; idx < D_HID * 16; idx += 256) {
            int k = idx >> 4, q = idx & 15;
            __builtin_amdgcn_global_load_async_to_lds_b128(
                (gv4i*)(W + (size_t)k * Dout + n0 + q * 4),
                (lv4i*)(&Wl[k * 64 + q * 4]),
                0, 0);
        }
        __builtin_amdgcn_s_wait_asynccnt(0);
    } else
#endif
    {
        for (int idx = tid; idx < D_HID * 64; idx += 256) {
            int k = idx >> 6, n = idx & 63, col = n0 + n;
            Wl[idx] = (col < Dout) ? W[(size_t)k * Dout + col] : 0.f;
        }
    }
    if (tid < 64) bl[tid] = (n0 + tid < Dout) ? bias[n0 + tid] : 0.f;
    __syncthreads();

    const int wave = tid >> 5;
    const int lane = tid & 31;
    const int half = lane >> 4;    // 0: K pair {0,1}, 1: K pair {2,3}
    const int l16  = lane & 15;

    const int mtiles = N >> 4;                        // N is a multiple of 16
    const int mt = blockIdx.x * 8 + wave;
    if (mt >= mtiles) return;                         // wave-uniform exit

    const float* Arow = A + (size_t)(mt * 16 + l16) * D_HID + 2 * half;

    v8f c[NT] = {};
    for (int k = 0; k < D_HID; k += 4) {
        float2 av = *(const float2*)(Arow + k);
        v2f a; a.x = av.x; a.y = av.y;
        const float* wp = &Wl[(k + 2 * half) * 64 + l16];
        v2f b[NT];
        #pragma unroll
        for (int nt = 0; nt < NT; ++nt) {
            b[nt].x = wp[nt * 16];
            b[nt].y = wp[nt * 16 + 64];
        }
        #pragma unroll
        for (int nt = 0; nt < NT; ++nt)
            c[nt] = __builtin_amdgcn_wmma_f32_16x16x4_f32(
                        /*neg_a=*/false, a, /*neg_b=*/false, b[nt],
                        /*c_mod=*/(short)0, c[nt],
                        /*reuse_a=*/false, /*reuse_b=*/false);
    }

    const int rbase = mt * 16 + half * 8;
    #pragma unroll
    for (int nt = 0; nt < NT; ++nt) {
        const int col = n0 + nt * 16 + l16;
        if (col < Dout) {
            const float bb = bl[nt * 16 + l16];
            #pragma unroll
            for (int r = 0; r < 8; ++r) {
                float v = c[nt][r] + bb;
                if (doRelu) v = fmaxf(v, 0.f);
                out[(size_t)(rbase + r) * Dout + col] = v;
            }
        }
    }
}

// ---------------------------------------------------------------------------
// 4) log-softmax + NLL per node. One wave per node (C = 40 <= 64).
// ---------------------------------------------------------------------------
__global__ void nll_kernel(const float* __restrict__ logits, // [N,C]
                           const int*   __restrict__ labels, // [N]
                           float* __restrict__ nll,          // [N]
                           int N, int C) {
    int wave = threadIdx.x >> 5;
    int lane = threadIdx.x & 31;
    int node = blockIdx.x * (blockDim.x >> 5) + wave;
    if (node >= N) return;
    const float* x = logits + (size_t)node * C;

    const float NEG = -3.4e38f;
    float x0 = (lane      < C) ? x[lane]      : NEG;
    float x1 = (lane + 32 < C) ? x[lane + 32] : NEG;
    float m = fmaxf(x0, x1);
    #pragma unroll
    for (int s = 16; s >= 1; s >>= 1)
        m = fmaxf(m, __shfl_xor(m, s, 32));

    float se = 0.f;
    if (lane      < C) se += expf(x0 - m);
    if (lane + 32 < C) se += expf(x1 - m);
    #pragma unroll
    for (int s = 16; s >= 1; s >>= 1)
        se += __shfl_xor(se, s, 32);

    if (lane == 0) {
        int lb = labels[node];
        float xl = x[lb];
        nll[node] = -(xl - m - logf(se));
    }
}

// ---------------------------------------------------------------------------
// 5) deterministic mean reduction: single block, fixed tree
// ---------------------------------------------------------------------------
__global__ void reduce_mean_kernel(const float* __restrict__ nll,
                                   float* __restrict__ out, int N) {
    __shared__ float sm[1024];
    int tid = threadIdx.x;
    float s = 0.f;
    for (int i = tid; i < N; i += 1024) s += nll[i];
    sm[tid] = s;
    __syncthreads();
    for (int st = 512; st > 0; st >>= 1) {
        if (tid < st) sm[tid] += sm[tid + st];
        __syncthreads();
    }
    if (tid == 0) out[0] = sm[0] / (float)N;
}

// ---------------------------------------------------------------------------
extern "C" void kernel_launch(void* const* d_in, const int* in_sizes, int n_in,
                              void* d_out, int out_size, void* d_ws, size_t ws_size,
                              hipStream_t stream) {
    const float* features = (const float*)d_in[0];
    const float* edge_w   = (const float*)d_in[1];
    const float* W1 = (const float*)d_in[2];
    const float* b1 = (const float*)d_in[3];
    const float* W2 = (const float*)d_in[4];
    const float* b2 = (const float*)d_in[5];
    const float* W3 = (const float*)d_in[6];
    const float* b3 = (const float*)d_in[7];
    const int* src    = (const int*)d_in[8];
    const int* dst    = (const int*)d_in[9];
    const int* labels = (const int*)d_in[10];

    const int N = in_sizes[0] / D_HID;   // 100000
    const int E = in_sizes[1];           // 1600000
    const int C = in_sizes[7];           // 40

    // workspace carve-out (all 16B aligned)
    char* ws = (char*)d_ws;
    size_t off = 0;
    float* agg    = (float*)(ws + off); off += (size_t)N * D_HID * sizeof(float);
    float* h      = (float*)(ws + off); off += (size_t)N * D_HID * sizeof(float);
    float* nll    = (float*)(ws + off); off += (size_t)N * sizeof(float);
    int*   rp     = (int*)  (ws + off); off += (size_t)(N + 4) * sizeof(int);
    float* logits = (float*)(ws + off);

    const int spmmBlocks = (N + 7) / 8;                 // 8 waves/block, 1 node/wave
    const int mtiles     = N / 16;                      // N % 16 == 0
    const dim3 gemmGridH((mtiles + 7) / 8, D_HID / 64); // 64-wide column strips
    const dim3 gemmGridC((mtiles + 7) / 8, (C + 63) / 64);

    rowptr_kernel<<<(N + 1 + 255) / 256, 256, 0, stream>>>(dst, E, rp, N);

    // layer 1: features -> agg -> h
    spmm_kernel<<<spmmBlocks, 256, 0, stream>>>((const float4*)features, edge_w, src, rp,
                                                (float4*)agg, N);
    gemm_wmma_kernel<<<gemmGridH, 256, 0, stream>>>(agg, W1, b1, h, N, D_HID, 1);

    // layer 2: h -> agg -> h  (gemm reads only agg, safe to overwrite h)
    spmm_kernel<<<spmmBlocks, 256, 0, stream>>>((const float4*)h, edge_w, src, rp,
                                                (float4*)agg, N);
    gemm_wmma_kernel<<<gemmGridH, 256, 0, stream>>>(agg, W2, b2, h, N, D_HID, 1);

    // layer 3: h -> agg -> logits [N,C]
    spmm_kernel<<<spmmBlocks, 256, 0, stream>>>((const float4*)h, edge_w, src, rp,
                                                (float4*)agg, N);
    gemm_wmma_kernel<<<gemmGridC, 256, 0, stream>>>(agg, W3, b3, logits, N, C, 0);

    // loss
    nll_kernel<<<spmmBlocks, 256, 0, stream>>>(logits, labels, nll, N, C);
    reduce_mean_kernel<<<1, 1024, 0, stream>>>(nll, (float*)d_out, N);
}